// GAT_81011673137252
// MI455X (gfx1250) — compile-verified
//
#include <hip/hip_runtime.h>
#include <math.h>

typedef float v2f __attribute__((ext_vector_type(2)));
typedef float v8f __attribute__((ext_vector_type(8)));

#define NN 100000
#define NEG_SLOPE 0.2f
#define EPSF 1e-16f

// ---------------------------------------------------------------- utilities
__global__ void fill_kernel(float* __restrict__ p, float v, int n) {
    int i = blockIdx.x * blockDim.x + threadIdx.x;
    if (i < n) p[i] = v;
}

__device__ __forceinline__ void atomicMaxF(float* addr, float v) {
    // float ordering matches int ordering for >=0, reversed-unsigned for <0
    if (v >= 0.0f) atomicMax((int*)addr, __float_as_int(v));
    else           atomicMin((unsigned int*)addr, (unsigned int)__float_as_int(v));
}

// ---------------------------------------------------------------- GEMM (WMMA fp32)
// H[nrows x 64] = X[nrows x K] @ W[K x 64], all row-major.
// One wave computes a 16x64 tile using V_WMMA_F32_16X16X4_F32 (K-step 4).
__global__ __launch_bounds__(128) void gemm_wmma_kernel(
        const float* __restrict__ X, const float* __restrict__ W,
        float* __restrict__ H, int nrows, int K) {
    int wave = blockIdx.x * (blockDim.x >> 5) + (threadIdx.x >> 5);
    int mtiles = nrows >> 4;
    if (wave >= mtiles) return;               // wave-uniform exit, EXEC stays full

    int lane  = threadIdx.x & 31;
    int r0    = lane & 15;                    // row (A) / col-in-tile (B,C,D)
    int khalf = lane >> 4;                    // 0: K=k0,k0+1  1: K=k0+2,k0+3

    const float* xrow = X + (size_t)(wave * 16 + r0) * (size_t)K;

    v8f c0 = {}, c1 = {}, c2 = {}, c3 = {};
    for (int k0 = 0; k0 < K; k0 += 4) {
        int ka = k0 + khalf * 2;
        v2f a;
        a.x = xrow[ka];
        a.y = xrow[ka + 1];

        const float* w0 = W + (size_t)ka * 64;        // row ka
        const float* w1 = w0 + 64;                    // row ka+1
        v2f b0, b1, b2, b3;
        b0.x = w0[r0];      b0.y = w1[r0];
        b1.x = w0[16 + r0]; b1.y = w1[16 + r0];
        b2.x = w0[32 + r0]; b2.y = w1[32 + r0];
        b3.x = w0[48 + r0]; b3.y = w1[48 + r0];

        c0 = __builtin_amdgcn_wmma_f32_16x16x4_f32(false, a, false, b0, (short)0, c0, false, false);
        c1 = __builtin_amdgcn_wmma_f32_16x16x4_f32(false, a, false, b1, (short)0, c1, false, false);
        c2 = __builtin_amdgcn_wmma_f32_16x16x4_f32(false, a, false, b2, (short)0, c2, false, false);
        c3 = __builtin_amdgcn_wmma_f32_16x16x4_f32(false, a, false, b3, (short)0, c3, false, false);
    }

    // D layout: VGPR r -> row (r + khalf*8), lanes%16 -> column within 16-tile
    float* hbase = H + (size_t)(wave * 16 + khalf * 8) * 64 + r0;
    #pragma unroll
    for (int r = 0; r < 8; ++r) {
        float* row = hbase + (size_t)r * 64;
        row[0]  = c0[r];
        row[16] = c1[r];
        row[32] = c2[r];
        row[48] = c3[r];
    }
}

// ---------------------------------------------------------------- per-node scores
// S[n] = dot(H[n,:], a_s), D[n] = dot(H[n,:], a_d); one wave per node.
__global__ __launch_bounds__(256) void scores_kernel(
        const float* __restrict__ H, const float* __restrict__ a_s,
        const float* __restrict__ a_d, float* __restrict__ S, float* __restrict__ D) {
    int node = blockIdx.x * (blockDim.x >> 5) + (threadIdx.x >> 5);
    if (node >= NN) return;
    int lane = threadIdx.x & 31;
    const float* h = H + (size_t)node * 64;
    float h0 = h[lane], h1 = h[lane + 32];
    float s = h0 * a_s[lane] + h1 * a_s[lane + 32];
    float d = h0 * a_d[lane] + h1 * a_d[lane + 32];
    #pragma unroll
    for (int off = 16; off; off >>= 1) {
        s += __shfl_xor(s, off, 32);
        d += __shfl_xor(d, off, 32);
    }
    if (lane == 0) { S[node] = s; D[node] = d; }
}

// ---------------------------------------------------------------- edge passes
__global__ void edge_max_kernel(const int* __restrict__ ei, int E, int Etot,
        const float* __restrict__ S, const float* __restrict__ D,
        float* __restrict__ M) {
    int i = blockIdx.x * blockDim.x + threadIdx.x;
    if (i >= Etot) return;
    int src, dst;
    if (i < E) { src = ei[i]; dst = ei[E + i]; } else { src = dst = i - E; }
    float a = S[src] + D[dst];
    a = (a >= 0.0f) ? a : NEG_SLOPE * a;
    atomicMaxF(&M[dst], a);
}

__global__ void edge_exp_kernel(const int* __restrict__ ei, int E, int Etot,
        const float* __restrict__ S, const float* __restrict__ D,
        const float* __restrict__ M, float* __restrict__ EW, float* __restrict__ Z) {
    int i = blockIdx.x * blockDim.x + threadIdx.x;
    if (i >= Etot) return;
    int src, dst;
    if (i < E) { src = ei[i]; dst = ei[E + i]; } else { src = dst = i - E; }
    float a = S[src] + D[dst];
    a = (a >= 0.0f) ? a : NEG_SLOPE * a;
    float e = expf(a - M[dst]);
    EW[i] = e;
    atomicAdd(&Z[dst], e);
}

// one wave per edge: 32 lanes x 2 channels = 64-ch weighted scatter-add
__global__ __launch_bounds__(256) void edge_agg_kernel(
        const int* __restrict__ ei, int E, int Etot,
        const float* __restrict__ H, const float* __restrict__ EW,
        const float* __restrict__ Z, float* __restrict__ ACC) {
    int e = blockIdx.x * 8 + (threadIdx.x >> 5);
    if (e >= Etot) return;
    int lane = threadIdx.x & 31;
    int src, dst;
    if (e < E) { src = ei[e]; dst = ei[E + e]; } else { src = dst = e - E; }
    float w = EW[e] / (Z[dst] + EPSF);
    const float* h = H + (size_t)src * 64;
    float* o = ACC + (size_t)dst * 64;
    atomicAdd(&o[lane],      h[lane]      * w);
    atomicAdd(&o[lane + 32], h[lane + 32] * w);
}

// ---------------------------------------------------------------- epilogues
__global__ void relu_bias_kernel(float* __restrict__ A, const float* __restrict__ b, int n) {
    int i = blockIdx.x * blockDim.x + threadIdx.x;
    if (i < n) {
        float v = A[i] + b[i & 63];
        A[i] = (v > 0.0f) ? v : 0.0f;
    }
}

__global__ void bias_kernel(float* __restrict__ A, const float* __restrict__ b, int n) {
    int i = blockIdx.x * blockDim.x + threadIdx.x;
    if (i < n) A[i] += b[i & 63];
}

// ---------------------------------------------------------------- launcher
extern "C" void kernel_launch(void* const* d_in, const int* in_sizes, int n_in,
                              void* d_out, int out_size, void* d_ws, size_t ws_size,
                              hipStream_t stream) {
    const float* x     = (const float*)d_in[0];
    const int*   ei    = (const int*)d_in[1];
    const float* W1    = (const float*)d_in[2];
    const float* as1   = (const float*)d_in[3];
    const float* ad1   = (const float*)d_in[4];
    const float* b1    = (const float*)d_in[5];
    const float* W2    = (const float*)d_in[6];
    const float* as2   = (const float*)d_in[7];
    const float* ad2   = (const float*)d_in[8];
    const float* b2    = (const float*)d_in[9];
    float* out = (float*)d_out;

    const int E    = in_sizes[1] / 2;
    const int Etot = E + NN;
    const int NF   = NN * 64;

    float* wsf = (float*)d_ws;
    float* H   = wsf;            // NN*64 : GEMM output (per layer)
    float* ACC = H   + NF;       // NN*64 : layer-1 aggregation / layer-2 input
    float* S   = ACC + NF;       // NN    : per-node src score
    float* D   = S   + NN;       // NN    : per-node dst score
    float* M   = D   + NN;       // NN    : segment max
    float* Z   = M   + NN;       // NN    : segment sum
    float* EW  = Z   + NN;       // Etot  : per-edge exp

    const int TB = 256;
    dim3 blk(TB);
    int gN    = (NN + TB - 1) / TB;
    int gNF   = (NF + TB - 1) / TB;
    int gE    = (Etot + TB - 1) / TB;
    int gEagg = (Etot + 7) / 8;              // 8 waves / block, wave per edge
    int gScr  = (NN + 7) / 8;                // 8 waves / block, wave per node
    int gGemm = ((NN / 16) + 3) / 4;         // 4 waves / block, wave per 16-row tile

    // ---------------- layer 1 ----------------
    fill_kernel<<<gNF, blk, 0, stream>>>(ACC, 0.0f, NF);
    fill_kernel<<<gN,  blk, 0, stream>>>(M, -INFINITY, NN);
    fill_kernel<<<gN,  blk, 0, stream>>>(Z, 0.0f, NN);

    gemm_wmma_kernel<<<gGemm, dim3(128), 0, stream>>>(x, W1, H, NN, 128);
    scores_kernel<<<gScr, blk, 0, stream>>>(H, as1, ad1, S, D);
    edge_max_kernel<<<gE, blk, 0, stream>>>(ei, E, Etot, S, D, M);
    edge_exp_kernel<<<gE, blk, 0, stream>>>(ei, E, Etot, S, D, M, EW, Z);
    edge_agg_kernel<<<gEagg, blk, 0, stream>>>(ei, E, Etot, H, EW, Z, ACC);
    relu_bias_kernel<<<gNF, blk, 0, stream>>>(ACC, b1, NF);   // ACC -> layer-2 input

    // ---------------- layer 2 ----------------
    fill_kernel<<<gN,  blk, 0, stream>>>(M, -INFINITY, NN);
    fill_kernel<<<gN,  blk, 0, stream>>>(Z, 0.0f, NN);
    fill_kernel<<<gNF, blk, 0, stream>>>(out, 0.0f, NF);

    gemm_wmma_kernel<<<gGemm, dim3(128), 0, stream>>>(ACC, W2, H, NN, 64);
    scores_kernel<<<gScr, blk, 0, stream>>>(H, as2, ad2, S, D);
    edge_max_kernel<<<gE, blk, 0, stream>>>(ei, E, Etot, S, D, M);
    edge_exp_kernel<<<gE, blk, 0, stream>>>(ei, E, Etot, S, D, M, EW, Z);
    edge_agg_kernel<<<gEagg, blk, 0, stream>>>(ei, E, Etot, H, EW, Z, out);
    bias_kernel<<<gNF, blk, 0, stream>>>(out, b2, NF);
}